// RoIAlign_15719580304123
// MI455X (gfx1250) — compile-verified
//
#include <hip/hip_runtime.h>

// RoI Align for MI455X (gfx1250, wave32).
// Gather-bandwidth bound: ~100M 4B gathers + 100MB stores vs ~0.2 GFLOP.
// The 4-tap bilinear combine is executed on the matrix pipe via
// V_WMMA_F32_16X16X4_F32 (K = 4 taps, N = 16 channels per issue, weights
// replicated across M rows), at exactly the same load count per output as a
// scalar formulation: 2 x b32 loads per lane per 16-channel tile.

typedef float v2f __attribute__((ext_vector_type(2)));
typedef float v8f __attribute__((ext_vector_type(8)));

namespace {
constexpr int   kC     = 256;
constexpr int   kH     = 200;
constexpr int   kW     = 304;
constexpr int   kOut   = 7;
constexpr float kScale = 0.25f;
}

__global__ __launch_bounds__(256) void roi_align_wmma(
    const float* __restrict__ feat,   // (4, 256, 200, 304) f32
    const float* __restrict__ rois,   // (N, 5) f32: b, x1, y1, x2, y2
    float* __restrict__ out)          // (N, 256, 7, 7) f32
{
    const int tid  = threadIdx.x;
    const int wave = tid >> 5;        // 0..7
    const int lane = tid & 31;
    const int half = lane >> 4;       // 0 -> taps on row y0, 1 -> taps on row y1
    const int n16  = lane & 15;       // WMMA N column == channel within chunk
    const int roi  = blockIdx.x;

    // ROI parameters (uniform over the block). ALIGNED=False -> offset 0.
    const float* r  = rois + roi * 5;
    const int   b   = (int)r[0];
    const float sx1 = r[1] * kScale;
    const float sy1 = r[2] * kScale;
    const float sx2 = r[3] * kScale;
    const float sy2 = r[4] * kScale;
    const float rw  = fmaxf(sx2 - sx1, 1.0f);
    const float rh  = fmaxf(sy2 - sy1, 1.0f);
    const float bw  = rw * (1.0f / (float)kOut);
    const float bh  = rh * (1.0f / (float)kOut);

    // Each wave owns channel chunks {wave, wave+8}; sweeps all 49 bins per
    // chunk so consecutive iterations stay inside the same 16 channel planes
    // (good L2 locality: bins of one ROI cluster spatially).
    for (int cc = wave; cc < 16; cc += 8) {
        const int c         = cc * 16 + n16;              // this lane's channel
        const int planeBase = ((b * kC + c) * kH) * kW;   // < 2^31, fits int

        for (int ph = 0; ph < kOut; ++ph) {
            // RATIO=1 -> single sample at bin center: grid = ph + 0.5
            const float y  = sy1 + ((float)ph + 0.5f) * bh;
            const bool  vy = (y > -1.0f) && (y < (float)kH);
            const float ycl = fminf(fmaxf(y, 0.0f), (float)(kH - 1));
            const int   y0  = (int)floorf(ycl);
            const int   y1i = min(y0 + 1, kH - 1);
            const float ly  = ycl - (float)y0;
            const float hy  = 1.0f - ly;
            // Lanes 0-15 read the y0 row pair, lanes 16-31 the y1 row pair.
            const int   yrow    = half ? y1i : y0;
            const int   rowBase = planeBase + yrow * kW;

            for (int pw = 0; pw < kOut; ++pw) {
                const float x     = sx1 + ((float)pw + 0.5f) * bw;
                const bool  valid = vy && (x > -1.0f) && (x < (float)kW);
                const float xcl = fminf(fmaxf(x, 0.0f), (float)(kW - 1));
                const int   x0  = (int)floorf(xcl);
                const int   x1i = min(x0 + 1, kW - 1);
                const float lx  = xcl - (float)x0;
                const float hx  = 1.0f - lx;

                // Tap order K0=(y0,x0) K1=(y0,x1) K2=(y1,x0) K3=(y1,x1)
                float w00 = hy * hx, w01 = hy * lx, w10 = ly * hx, w11 = ly * lx;
                w00 = valid ? w00 : 0.0f;
                w01 = valid ? w01 : 0.0f;
                w10 = valid ? w10 : 0.0f;
                w11 = valid ? w11 : 0.0f;

                // Two b32 gathers per lane; x0/x1 adjacent -> usually one line.
                const float fa = feat[rowBase + x0];
                const float fb = feat[rowBase + x1i];

                // A (16x4 f32, ISA layout): VGPR0 = K0 | K2 per lane-half,
                // VGPR1 = K1 | K3. Weights replicated over all M rows.
                v2f A;
                A.x = half ? w10 : w00;
                A.y = half ? w11 : w01;
                // B (4x16 f32): row-striped across lanes, rows k and k+2 in
                // the two lane-halves of each VGPR -> B.x = row-y tap at x0,
                // B.y = row-y tap at x1 for every lane.
                v2f Bm;
                Bm.x = fa;
                Bm.y = fb;
                v8f Cacc = {};
                // D[m, n] = w00*F00[n] + w01*F01[n] + w10*F10[n] + w11*F11[n]
                // identical for every row m -> D[0] in each lane is the result
                // for channel (lane & 15).
                v8f D = __builtin_amdgcn_wmma_f32_16x16x4_f32(
                    /*neg_a=*/false, A, /*neg_b=*/false, Bm,
                    /*c_mod=*/(short)0, Cacc,
                    /*reuse_a=*/false, /*reuse_b=*/false);

                if (lane < 16) {
                    out[(roi * kC + c) * (kOut * kOut) + ph * kOut + pw] = D[0];
                }
            }
        }
    }
}

extern "C" void kernel_launch(void* const* d_in, const int* in_sizes, int n_in,
                              void* d_out, int out_size, void* d_ws, size_t ws_size,
                              hipStream_t stream) {
    (void)n_in; (void)out_size; (void)d_ws; (void)ws_size;
    const float* feat = (const float*)d_in[0];
    const float* rois = (const float*)d_in[1];
    float*       out  = (float*)d_out;
    const int N = in_sizes[1] / 5;   // 2000 ROIs

    // One block per ROI, 256 threads = 8 waves; each wave handles one bin per
    // WMMA issue across 16 channels.
    roi_align_wmma<<<dim3(N), dim3(256), 0, stream>>>(feat, rois, out);
}